// GIN_60559038874094
// MI455X (gfx1250) — compile-verified
//
#include <hip/hip_runtime.h>
#include <cstdint>

typedef __attribute__((ext_vector_type(2))) float v2f;
typedef __attribute__((ext_vector_type(8))) float v8f;

// ---------------------------------------------------------------------------
// Zero-fill helper (agg scratch + output accumulator are 0xAA-poisoned)
// ---------------------------------------------------------------------------
__global__ void zero_f32(float* __restrict__ p, size_t n) {
    size_t i = (size_t)blockIdx.x * blockDim.x + threadIdx.x;
    if (i < n) p[i] = 0.0f;
}

// ---------------------------------------------------------------------------
// Edge scatter: agg[dst] += x[src].
// One edge per 16-lane half-wave: each lane loads a float4 (global_load_b128)
// slice of the 64-wide row and issues 4 non-returning f32 atomics.
// The ~102M dword atomics are the irreducible L2-RMW floor of this op; the
// b128 loads + halved index traffic minimize everything around them.
// Both x (25.6MB) and agg (25.6MB) are L2-resident (192MB L2).
// ---------------------------------------------------------------------------
__global__ __launch_bounds__(256) void gin_scatter(
    const float* __restrict__ x,
    const int*   __restrict__ src,
    const int*   __restrict__ dst,
    float*       __restrict__ agg,
    int n_edges)
{
    const size_t t = (size_t)blockIdx.x * blockDim.x + threadIdx.x;
    const int e = (int)(t >> 4);            // one edge per 16 lanes
    if (e >= n_edges) return;
    const int col = ((int)t & 15) << 2;     // 4-float slice

    const int s = src[e];
    const int d = dst[e];

    const float4 v = *(const float4*)(x + (size_t)s * 64 + col);
    float* dp = agg + (size_t)d * 64 + col;
    atomicAdd(dp + 0, v.x);
    atomicAdd(dp + 1, v.y);
    atomicAdd(dp + 2, v.z);
    atomicAdd(dp + 3, v.w);
}

// ---------------------------------------------------------------------------
// Fused: h0 = x + agg ; h1 = relu(h0@W1 + b1) ; h2 = h1@W2 + b2 ;
//        out += sum_i h2[i,:] * w[i]
// One wave per 16-node tile, 4 waves / block, fp32 WMMA 16x16x4.
//
// ISA VGPR layouts (CDNA5 ISA §7.12.2):
//   A (16x4 f32):  lane L holds row M=L%16; VGPR0/1 = K pair (L/16)*2, +1
//   B (4x16 f32):  lane L holds col N=L%16; VGPR0/1 = K pair (L/16)*2, +1
//   C/D (16x16):   VGPR r, lane L -> row r + (L/16)*8, col L%16
//
// W1/W2 are staged TRANSPOSED into LDS once per block so each lane's B
// operand K-pair is contiguous -> single ds_load_b64 (pitch 66 keeps the
// dword index even => 8B-aligned, and stride-2 banking is conflict-free).
// Output is accumulated per-block in LDS (ds_add_f32) and flushed with one
// 64-wide global atomic per block instead of 400k contended atomics.
// ---------------------------------------------------------------------------
#define WAVES_PER_BLOCK 4
#define H_PITCH 68   // 16-row h tiles: 64 + pad (even => b64-aligned)
#define W_PITCH 66   // transposed weights: 64 + pad (even => b64-aligned)

__global__ __launch_bounds__(WAVES_PER_BLOCK * 32) void gin_mlp(
    const float* __restrict__ x,
    const float* __restrict__ agg,
    const float* __restrict__ wts,
    const float* __restrict__ W1, const float* __restrict__ b1,
    const float* __restrict__ W2, const float* __restrict__ b2,
    float* __restrict__ out,
    int n_tiles)
{
    __shared__ float lds_h0[WAVES_PER_BLOCK][16][H_PITCH];
    __shared__ float lds_h1[WAVES_PER_BLOCK][16][H_PITCH];
    __shared__ float w1t[64][W_PITCH];   // w1t[n][k] = W1[k][n]
    __shared__ float w2t[64][W_PITCH];
    __shared__ float outacc[64];

    const int wave = threadIdx.x >> 5;
    const int lane = threadIdx.x & 31;
    const int tile = blockIdx.x * WAVES_PER_BLOCK + wave;
    const int base = tile * 16;

    // --- block-cooperative: transpose-stage W1/W2, zero output scratch ---
    for (int i = threadIdx.x; i < 4096; i += WAVES_PER_BLOCK * 32) {
        const int k = i >> 6, n = i & 63;
        w1t[n][k] = W1[i];
        w2t[n][k] = W2[i];
    }
    if (threadIdx.x < 64) outacc[threadIdx.x] = 0.0f;

    float (*h0)[H_PITCH] = lds_h0[wave];
    float (*h1)[H_PITCH] = lds_h1[wave];

    // --- per-wave: stage h0 = x + agg (16 rows x 64 cols, float4 loads) ---
    if (tile < n_tiles) {
        for (int i = lane; i < 256; i += 32) {
            const int row = i >> 4;
            const int c4  = (i & 15) << 2;
            const size_t g = (size_t)(base + row) * 64 + c4;
            const float4 xv = *(const float4*)(x + g);
            const float4 av = *(const float4*)(agg + g);
            float4 hv;
            hv.x = xv.x + av.x; hv.y = xv.y + av.y;
            hv.z = xv.z + av.z; hv.w = xv.w + av.w;
            *(float4*)&h0[row][c4] = hv;
        }
    }
    __syncthreads();
    if (tile >= n_tiles) return;   // whole-wave exit -> EXEC stays all-1s

    const int colN  = lane & 15;          // A: row M; B/C/D: col N
    const int khalf = (lane >> 4) << 1;   // K-pair select (0 or 2)
    const int rowhi = (lane >> 4) << 3;   // C/D row offset (0 or 8)

    // --- layer 1: h1 = relu(h0 @ W1 + b1) ---
    for (int nt = 0; nt < 4; ++nt) {
        const int n0 = nt << 4;
        const float* brow = &w1t[n0 + colN][0];
        v8f c;
        const float bias = b1[n0 + colN];
        #pragma unroll
        for (int r = 0; r < 8; ++r) c[r] = bias;   // bias folded into C

        #pragma unroll
        for (int k0 = 0; k0 < 64; k0 += 4) {
            const int ka = k0 + khalf;
            const v2f a = *(const v2f*)&h0[colN][ka];   // ds_load_b64
            const v2f b = *(const v2f*)&brow[ka];       // ds_load_b64
            c = __builtin_amdgcn_wmma_f32_16x16x4_f32(
                    false, a, false, b, (short)0, c, false, false);
        }
        #pragma unroll
        for (int r = 0; r < 8; ++r) {
            const float v = c[r];
            h1[r + rowhi][n0 + colN] = v > 0.0f ? v : 0.0f;   // ReLU
        }
    }
    __syncthreads();

    // --- layer 2 + weighted node reduction ---
    float wrow[8];
    #pragma unroll
    for (int r = 0; r < 8; ++r) wrow[r] = wts[base + r + rowhi];

    for (int nt = 0; nt < 4; ++nt) {
        const int n0 = nt << 4;
        const float* brow = &w2t[n0 + colN][0];
        v8f c;
        const float bias = b2[n0 + colN];
        #pragma unroll
        for (int r = 0; r < 8; ++r) c[r] = bias;

        #pragma unroll
        for (int k0 = 0; k0 < 64; k0 += 4) {
            const int ka = k0 + khalf;
            const v2f a = *(const v2f*)&h1[colN][ka];
            const v2f b = *(const v2f*)&brow[ka];
            c = __builtin_amdgcn_wmma_f32_16x16x4_f32(
                    false, a, false, b, (short)0, c, false, false);
        }
        // per-lane weighted sum over its 8 rows, fold rows 8..15 in, then
        // accumulate into block-local LDS scratch (ds_add_f32)
        float p = 0.0f;
        #pragma unroll
        for (int r = 0; r < 8; ++r) p += c[r] * wrow[r];
        p += __shfl_xor(p, 16, 32);
        if (lane < 16) atomicAdd(&outacc[n0 + colN], p);
    }
    __syncthreads();

    // one 64-wide global atomic flush per block, from wave 0 (always valid)
    if (threadIdx.x < 32) {
        for (int c = lane; c < 64; c += 32)
            atomicAdd(out + c, outacc[c]);
    }
}

// ---------------------------------------------------------------------------
extern "C" void kernel_launch(void* const* d_in, const int* in_sizes, int n_in,
                              void* d_out, int out_size, void* d_ws, size_t ws_size,
                              hipStream_t stream) {
    const float* x   = (const float*)d_in[0];
    const int*   ei  = (const int*)d_in[1];   // edge_index [2, E]
    const float* wts = (const float*)d_in[2];
    const float* W1  = (const float*)d_in[3];
    const float* b1  = (const float*)d_in[4];
    const float* W2  = (const float*)d_in[5];
    const float* b2  = (const float*)d_in[6];
    float* out = (float*)d_out;

    const int n_nodes = in_sizes[0] / 64;
    const int n_edges = in_sizes[1] / 2;
    const int n_tiles = n_nodes / 16;         // 100000/16 = 6250

    float* agg = (float*)d_ws;                // n_nodes*64 f32 = 25.6 MB
    const size_t agg_n = (size_t)n_nodes * 64;

    // 1) zero agg + out
    zero_f32<<<(unsigned)((agg_n + 255) / 256), 256, 0, stream>>>(agg, agg_n);
    zero_f32<<<1, 64, 0, stream>>>(out, (size_t)out_size);

    // 2) edge scatter (one edge per 16-lane half-wave, b128 loads)
    {
        const size_t threads = (size_t)n_edges * 16;
        gin_scatter<<<(unsigned)((threads + 255) / 256), 256, 0, stream>>>(
            x, ei, ei + n_edges, agg, n_edges);
    }

    // 3) fused MLP + weighted reduce (one wave per 16-node tile)
    {
        const int blocks = (n_tiles + WAVES_PER_BLOCK - 1) / WAVES_PER_BLOCK;
        gin_mlp<<<blocks, WAVES_PER_BLOCK * 32, 0, stream>>>(
            x, agg, wts, W1, b1, W2, b2, out, n_tiles);
    }
}